// GCN_node_classification_69277822484503
// MI455X (gfx1250) — compile-verified
//
#include <hip/hip_runtime.h>
#include <hip/hip_bf16.h>
#include <math.h>

// ---------------------------------------------------------------------------
// GCN node classification on MI455X (gfx1250), f32 WMMA (V_WMMA_F32_16X16X4_F32)
//
// gso is never materialized:
//   gso @ Y = rowscale ⊙ (adj @ (colscale⊙Y)) + rowscale·d1 ⊙ (colscale⊙Y)
//             + diagterm ⊙ Y
// rowscale = c2*diags^q, colscale = diags^r, diagterm = c1*diags^p + c3,
// diags = rowsum(adj) + d1.
// HBM floor: adj (268MB) read 3x ≈ 35us @ 23.3 TB/s; f32 WMMA keeps exact
// reference precision while using the matrix pipe.
// ---------------------------------------------------------------------------

#define GN 8192

typedef float f32x2 __attribute__((ext_vector_type(2)));
typedef float f32x8 __attribute__((ext_vector_type(8)));

#if defined(__HIP_DEVICE_COMPILE__) && defined(__has_builtin)
#if __has_builtin(__builtin_amdgcn_global_load_async_to_lds_b128)
#define ASYNC_LDS 1
#endif
#if __has_builtin(__builtin_amdgcn_s_wait_asynccnt)
#define HAVE_WAIT_ASYNC 1
#endif
#endif

// Pointer types for the async-to-LDS builtin: pointee must be the GCC-style
// 16B int vector (per hipcc diagnostic), global-AS src / LDS-AS dst.
typedef int v4i __attribute__((vector_size(16)));
typedef __attribute__((address_space(1))) v4i g_v4i;
typedef __attribute__((address_space(3))) v4i l_v4i;

__device__ __forceinline__ f32x8 wmma4(f32x2 a, f32x2 b, f32x8 c) {
  // D(16x16,f32) = A(16x4,f32) * B(4x16,f32) + C
  return __builtin_amdgcn_wmma_f32_16x16x4_f32(
      /*neg_a=*/false, a, /*neg_b=*/false, b,
      /*c_mod=*/(short)0, c, /*reuse_a=*/false, /*reuse_b=*/false);
}

// ---------------------------------------------------------------------------
// Kernel 1: per-row scales from adj row sums. One wave per row.
// ---------------------------------------------------------------------------
__global__ __launch_bounds__(256)
void k_rowprep(const float* __restrict__ adj, const float* __restrict__ d1,
               const float* __restrict__ p, const float* __restrict__ q,
               const float* __restrict__ r, const float* __restrict__ c1,
               const float* __restrict__ c2, const float* __restrict__ c3,
               float* __restrict__ rowscale, float* __restrict__ colscale,
               float* __restrict__ diagterm) {
  int row  = blockIdx.x * 8 + (threadIdx.x >> 5);
  int lane = threadIdx.x & 31;
  const float4* a4 = (const float4*)(adj + (size_t)row * GN);
  float s = 0.f;
  for (int i = lane; i < GN / 4; i += 32) {
    float4 v = a4[i];
    s += v.x + v.y + v.z + v.w;
  }
  for (int off = 16; off > 0; off >>= 1) s += __shfl_xor(s, off, 32);
  if (lane == 0) {
    float dg = s + d1[row];
    rowscale[row] = c2[0] * powf(dg, q[0]);
    colscale[row] = powf(dg, r[0]);
    diagterm[row] = c1[0] * powf(dg, p[0]) + c3[0];
  }
}

// ---------------------------------------------------------------------------
// Kernel 2/5: small feature GEMM  Y = A[GN][K] @ B[K][N], Ys = colscale⊙Y.
// 128 threads = 4 waves; wave w computes rows [blk*64 + 16w, +16) x all N.
// ---------------------------------------------------------------------------
template <int K, int N>
__global__ __launch_bounds__(128)
void k_gemm_small(const float* __restrict__ A, const float* __restrict__ B,
                  const float* __restrict__ colscale,
                  float* __restrict__ Y, float* __restrict__ Ys) {
  constexpr int NT = N / 16;
  int wave = threadIdx.x >> 5;
  int lane = threadIdx.x & 31;
  int half = lane >> 4;   // 0: K+{0,1}, 1: K+{2,3}
  int l16  = lane & 15;
  int m0   = blockIdx.x * 64 + wave * 16;

  const f32x8 zero = {0.f, 0.f, 0.f, 0.f, 0.f, 0.f, 0.f, 0.f};
  f32x8 acc[NT];
#pragma unroll
  for (int t = 0; t < NT; ++t) acc[t] = zero;

  const float* arow = A + (size_t)(m0 + l16) * K + 2 * half;
  for (int k = 0; k < K; k += 4) {
    f32x2 af = *(const f32x2*)(arow + k);
#pragma unroll
    for (int t = 0; t < NT; ++t) {
      f32x2 bf;
      bf.x = B[(size_t)(k + 2 * half) * N + t * 16 + l16];
      bf.y = B[(size_t)(k + 2 * half + 1) * N + t * 16 + l16];
      acc[t] = wmma4(af, bf, acc[t]);
    }
  }
#pragma unroll
  for (int t = 0; t < NT; ++t) {
#pragma unroll
    for (int v = 0; v < 8; ++v) {
      int row = m0 + v + 8 * half;
      size_t idx = (size_t)row * N + t * 16 + l16;
      float y = acc[t][v];
      Y[idx]  = y;
      Ys[idx] = y * colscale[row];
    }
  }
}

// ---------------------------------------------------------------------------
// Kernel 3/6: big GEMM  part[split] = adj[:, ksplit] @ Bm[ksplit, :N].
// 128 threads = 4 waves; block tile 128(M) x N; each wave owns 32 rows so
// every B fragment feeds 2 WMMAs. B is pair-interleaved in LDS so a B
// fragment is a single ds_load_b64. A tile staged with async-to-LDS when
// the builtin exists (ASYNCcnt path), float4 fallback otherwise.
// ---------------------------------------------------------------------------
template <int N, int KSPLIT>
__global__ __launch_bounds__(128)
void k_gemm_adj(const float* __restrict__ adj, const float* __restrict__ Bm,
                float* __restrict__ part) {
  constexpr int KC   = 32;
  constexpr int BM   = 128;
  constexpr int NT   = N / 16;
  constexpr int KPER = GN / KSPLIT;
  constexpr int RP   = 2 * N + 8;       // pair-row stride (floats), 8B-aligned
  __shared__ float Alds[BM][KC + 4];    // row stride 36 floats = 144B (16B-aligned)
  __shared__ float BldsP[(KC / 2) * RP];

  int m0    = blockIdx.x * BM;
  int split = blockIdx.y;
  int k0    = split * KPER;
  int tid   = threadIdx.x;
  int wave  = tid >> 5;
  int lane  = tid & 31;
  int half  = lane >> 4;
  int l16   = lane & 15;

  const f32x8 zero = {0.f, 0.f, 0.f, 0.f, 0.f, 0.f, 0.f, 0.f};
  f32x8 acc0[NT], acc1[NT];
#pragma unroll
  for (int t = 0; t < NT; ++t) { acc0[t] = zero; acc1[t] = zero; }

  int arow0 = wave * 32 + l16;
  int arow1 = arow0 + 16;

  for (int kb = 0; kb < KPER; kb += KC) {
    int kg = k0 + kb;
    // ---- Stage A: adj[m0..m0+127][kg..kg+31] (16KB), 8 x b128 per thread.
#pragma unroll
    for (int i = 0; i < (BM * KC / 4) / 128; ++i) {
      int c  = i * 128 + tid;
      int r  = c >> 3;
      int cc = (c & 7) * 4;
      const float* gp = adj + (size_t)(m0 + r) * GN + kg + cc;
#ifdef ASYNC_LDS
      __builtin_amdgcn_global_load_async_to_lds_b128(
          (g_v4i*)gp, (l_v4i*)&Alds[r][cc], 0, 0);
#else
      *(float4*)&Alds[r][cc] = *(const float4*)gp;
#endif
    }
    // ---- Stage B: Bm[kg..kg+31][0..N-1], pair-interleaved scatter.
    {
      const float4* src = (const float4*)(Bm + (size_t)kg * N);
#pragma unroll
      for (int e4 = 0; e4 < (KC * N / 4) / 128; ++e4) {
        int e = e4 * 128 + tid;
        float4 v = src[e];
        int k = (e * 4) / N;
        int n = (e * 4) % N;
        float* dst = &BldsP[(k >> 1) * RP + 2 * n + (k & 1)];
        dst[0] = v.x; dst[2] = v.y; dst[4] = v.z; dst[6] = v.w;
      }
    }
#ifdef ASYNC_LDS
#ifdef HAVE_WAIT_ASYNC
    __builtin_amdgcn_s_wait_asynccnt(0);
#else
    asm volatile("s_wait_asynccnt 0x0" ::: "memory");
#endif
#endif
    __syncthreads();
    // ---- Compute: per K=4 step: 2 A b64 + NT B b64 -> 2*NT WMMA.
#pragma unroll
    for (int kk = 0; kk < KC; kk += 4) {
      int kx = kk + 2 * half;  // even
      f32x2 a0 = *(const f32x2*)&Alds[arow0][kx];
      f32x2 a1 = *(const f32x2*)&Alds[arow1][kx];
      const float* bbase = &BldsP[(kx >> 1) * RP + 2 * l16];
#pragma unroll
      for (int t = 0; t < NT; ++t) {
        f32x2 bf = *(const f32x2*)(bbase + 32 * t);  // (B[kx][n], B[kx+1][n])
        acc0[t] = wmma4(a0, bf, acc0[t]);
        acc1[t] = wmma4(a1, bf, acc1[t]);
      }
    }
    __syncthreads();
  }
  // ---- Store partial tiles.
#pragma unroll
  for (int t = 0; t < NT; ++t) {
#pragma unroll
    for (int v = 0; v < 8; ++v) {
      int r0 = m0 + wave * 32 + v + 8 * half;
      size_t base = ((size_t)split * GN + r0) * N + t * 16 + l16;
      part[base] = acc0[t][v];
      part[base + (size_t)16 * N] = acc1[t][v];
    }
  }
}

// ---------------------------------------------------------------------------
// Kernel 4: reduce K-split partials + laplacian epilogue + bias + relu (f4).
// ---------------------------------------------------------------------------
template <int KSPLIT>
__global__ __launch_bounds__(256)
void k_epilogue1(const float* __restrict__ part, const float* __restrict__ Y0,
                 const float* __restrict__ Ys0, const float* __restrict__ rowscale,
                 const float* __restrict__ diagterm, const float* __restrict__ d1,
                 const float* __restrict__ b0, float* __restrict__ H1) {
  int i4  = blockIdx.x * 256 + threadIdx.x;  // over GN*128/4
  int row = i4 >> 5;
  int col = (i4 & 31) * 4;
  float4 s = {0.f, 0.f, 0.f, 0.f};
#pragma unroll
  for (int sp = 0; sp < KSPLIT; ++sp) {
    float4 v = *(const float4*)&part[((size_t)sp * GN + row) * 128 + col];
    s.x += v.x; s.y += v.y; s.z += v.z; s.w += v.w;
  }
  size_t idx = (size_t)row * 128 + col;
  float4 y0 = *(const float4*)&Y0[idx];
  float4 ys = *(const float4*)&Ys0[idx];
  float4 bb = *(const float4*)&b0[col];
  float rs = rowscale[row], dt = diagterm[row], dd = d1[row];
  float4 h;
  h.x = fmaxf(rs * (s.x + dd * ys.x) + dt * y0.x + bb.x, 0.f);
  h.y = fmaxf(rs * (s.y + dd * ys.y) + dt * y0.y + bb.y, 0.f);
  h.z = fmaxf(rs * (s.z + dd * ys.z) + dt * y0.z + bb.z, 0.f);
  h.w = fmaxf(rs * (s.w + dd * ys.w) + dt * y0.w + bb.w, 0.f);
  *(float4*)&H1[idx] = h;
}

// ---------------------------------------------------------------------------
// Kernel 7: reduce partials + epilogue + fused log_softmax (row of 64).
// One wave per row; lane handles columns {lane, lane+32}.
// ---------------------------------------------------------------------------
template <int KSPLIT>
__global__ __launch_bounds__(256)
void k_epilogue2(const float* __restrict__ part, const float* __restrict__ Y1,
                 const float* __restrict__ Ys1, const float* __restrict__ rowscale,
                 const float* __restrict__ diagterm, const float* __restrict__ d1,
                 const float* __restrict__ b1, float* __restrict__ out) {
  int row  = blockIdx.x * 8 + (threadIdx.x >> 5);
  int lane = threadIdx.x & 31;
  float rs = rowscale[row], dt = diagterm[row], dd = d1[row];
  float h[2];
#pragma unroll
  for (int j = 0; j < 2; ++j) {
    int col = lane + 32 * j;
    size_t idx = (size_t)row * 64 + col;
    float s = 0.f;
#pragma unroll
    for (int sp = 0; sp < KSPLIT; ++sp)
      s += part[((size_t)sp * GN + row) * 64 + col];
    h[j] = rs * (s + dd * Ys1[idx]) + dt * Y1[idx] + b1[col];
  }
  float m = fmaxf(h[0], h[1]);
  for (int off = 16; off > 0; off >>= 1) m = fmaxf(m, __shfl_xor(m, off, 32));
  float e = expf(h[0] - m) + expf(h[1] - m);
  for (int off = 16; off > 0; off >>= 1) e += __shfl_xor(e, off, 32);
  float lse = m + logf(e);
#pragma unroll
  for (int j = 0; j < 2; ++j)
    out[(size_t)row * 64 + lane + 32 * j] = h[j] - lse;
}

// ---------------------------------------------------------------------------
extern "C" void kernel_launch(void* const* d_in, const int* in_sizes, int n_in,
                              void* d_out, int out_size, void* d_ws, size_t ws_size,
                              hipStream_t stream) {
  const float* x   = (const float*)d_in[0];
  const float* adj = (const float*)d_in[1];
  const float* W0  = (const float*)d_in[2];
  const float* b0  = (const float*)d_in[3];
  const float* W1  = (const float*)d_in[4];
  const float* b1  = (const float*)d_in[5];
  const float* p   = (const float*)d_in[6];
  const float* q   = (const float*)d_in[7];
  const float* r   = (const float*)d_in[8];
  const float* c1  = (const float*)d_in[9];
  const float* c2  = (const float*)d_in[10];
  const float* c3  = (const float*)d_in[11];
  const float* d1  = (const float*)d_in[12];
  float* out = (float*)d_out;
  float* ws  = (float*)d_ws;

  constexpr int KS = 8;

  // Workspace layout (floats); part region reused by both adj-GEMMs. ~48MB.
  float* rowscale = ws;
  float* colscale = rowscale + GN;
  float* diagterm = colscale + GN;
  float* Y0   = diagterm + GN;          // GN*128
  float* Ys0  = Y0  + (size_t)GN * 128; // GN*128
  float* H1   = Ys0 + (size_t)GN * 128; // GN*128
  float* Y1   = H1  + (size_t)GN * 128; // GN*64
  float* Ys1  = Y1  + (size_t)GN * 64;  // GN*64
  float* part = Ys1 + (size_t)GN * 64;  // KS*GN*128 (>= KS*GN*64)

  k_rowprep<<<GN / 8, 256, 0, stream>>>(adj, d1, p, q, r, c1, c2, c3,
                                        rowscale, colscale, diagterm);
  k_gemm_small<256, 128><<<GN / 64, 128, 0, stream>>>(x, W0, colscale, Y0, Ys0);
  k_gemm_adj<128, KS><<<dim3(GN / 128, KS), 128, 0, stream>>>(adj, Ys0, part);
  k_epilogue1<KS><<<(GN * 128 / 4) / 256, 256, 0, stream>>>(part, Y0, Ys0, rowscale,
                                                            diagterm, d1, b0, H1);
  k_gemm_small<128, 64><<<GN / 64, 128, 0, stream>>>(H1, W1, colscale, Y1, Ys1);
  k_gemm_adj<64, KS><<<dim3(GN / 128, KS), 128, 0, stream>>>(adj, Ys1, part);
  k_epilogue2<KS><<<GN / 8, 256, 0, stream>>>(part, Y1, Ys1, rowscale, diagterm,
                                              d1, b1, out);
}